// LocallyConnectedLayer_79285096284188
// MI455X (gfx1250) — compile-verified
//
#include <hip/hip_runtime.h>

typedef __attribute__((ext_vector_type(2))) float v2f;
typedef __attribute__((ext_vector_type(8))) float v8f;
typedef __attribute__((ext_vector_type(4))) unsigned int u32x4;
typedef __attribute__((ext_vector_type(8))) int i32x8;
typedef __attribute__((ext_vector_type(4))) int i32x4;

#define B_   32   // batch
#define C_   32   // in channels
#define O_   64   // out channels
#define H_   64
#define W_   64
#define KK   3
#define OH_  62
#define OW_  62
#define CC   4             // input channels staged per K-chunk
#define KC   (CC*KK*KK)    // 36 k-values per chunk
#define TJ   8             // output columns per workgroup (one wave each)
#define XCOL (TJ + KK - 1) // 10 x-columns staged
#define NSTEP (KC/4)       // 9 WMMA K-steps per chunk
#define XBSTR (CC*3*XCOL)  // per-batch-row stride in Xl = 120
#define ROWE  72           // weight elements per (o,cl) row: 8 j * 9 uv
#define ROWW  81           // padded LDS row: 72 + 72/8 pad dwords (TDM pad 1 DW per 8)
#define CSTR  (OH_*OW_*9)  // weight c-stride in elements = 34596
#define OSTR  (C_*OH_*OW_*9) // weight o-stride = 1107072

// Padded LDS index for weight element (row r = o*CC+cl, in-row element e)
__device__ __forceinline__ int widx(int r, int e) {
  return r*ROWW + e + (e >> 3);
}

__global__ __launch_bounds__(256)
void lcl_wmma_kernel(const float* __restrict__ Xg, const float* __restrict__ Wg,
                     const float* __restrict__ Bg, float* __restrict__ Og) {
  extern __shared__ float smem[];
  float* Wl = smem;                   // [o*CC+cl][81] padded rows: 20736 floats
  float* Xl = smem + O_*CC*ROWW;      // [b][cl][u][col]: 3840 floats

  const int i  = blockIdx.y;
  const int j0 = blockIdx.x * TJ;
  const int validj = (OW_ - j0 < TJ) ? (OW_ - j0) : TJ;

  const int tid  = threadIdx.x;
  const int lane = tid & 31;
  const int wid  = tid >> 5;     // wave id == local j position
  const int half = lane >> 4;
  const int row  = lane & 15;

  // Block-uniform tile start inside weights (o=0, c=0): advances by CC*CSTR per chunk
  const unsigned long long gbase =
      (unsigned long long)(uintptr_t)(Wg + (size_t)(i*OW_ + j0) * 9);
  const unsigned int ldsW = (unsigned int)(uintptr_t)Wl;  // LDS byte offset (low 32 bits of flat)

  v8f acc[2][4];
  #pragma unroll
  for (int m = 0; m < 2; ++m)
    #pragma unroll
    for (int n = 0; n < 4; ++n)
      acc[m][n] = (v8f){0.f,0.f,0.f,0.f,0.f,0.f,0.f,0.f};

  for (int cc = 0; cc < C_/CC; ++cc) {
    const int cbase = cc * CC;
    __syncthreads();   // previous chunk's compute done before LDS overwrite

#if __has_builtin(__builtin_amdgcn_tensor_load_to_lds)
    // ---- TDM: DMA the 3D weight tile [o=64][cl=4][72 contig floats] -> LDS,
    //      zero-filling the ragged j tail (tensor_dim0 = validj*9 < tile_dim0),
    //      with 1-DWORD LDS pad per 8 DWORDs (row stride 81 -> bank-conflict-free).
    if (wid == 0) {
      const unsigned long long ga = gbase + (unsigned long long)cbase * (CSTR * 4ull);
      const unsigned int td0 = (unsigned int)(validj * 9);
      u32x4 g0 = { 1u,                                  // count=1 (valid user D#)
                   ldsW,                                // lds_addr (bytes)
                   (unsigned int)ga,                    // global_addr[31:0]
                   (unsigned int)(ga >> 32) | 0x80000000u }; // addr[56:32] | type=2<<30
      i32x8 g1 = { (int)((2u<<16) | (1u<<20) | (2u<<22)), // data_size=4B | pad_enable | pad_interval=8DW (amount=1DW)
                   (int)(td0 << 16),                      // tensor_dim0[15:0] (OOB->zero past validj)
                   (int)(32u << 16),                      // tensor_dim0 hi=0 | tensor_dim1=32
                   (int)((unsigned)ROWE << 16),           // tensor_dim1 hi=0 | tile_dim0=72
                   (int)(4u | (64u << 16)),               // tile_dim1=4 | tile_dim2=64
                   (int)CSTR,                             // tensor_dim0_stride lo32
                   (int)((OSTR & 0xFFFFu) << 16),         // stride0 hi=0 | tensor_dim1_stride[15:0]
                   (int)(OSTR >> 16) };                   // tensor_dim1_stride[47:16]
      i32x4 g2 = { 64, 0, 0, 0 };                         // tensor_dim2=64; dim3 unused
      i32x4 g3 = { 0, 0, 0, 0 };
#if __has_include(<hip/amd_detail/amd_gfx1250_TDM.h>)
      i32x8 g4 = { 0,0,0,0,0,0,0,0 };
      __builtin_amdgcn_tensor_load_to_lds(g0, g1, g2, g3, g4, 0);
#else
      __builtin_amdgcn_tensor_load_to_lds(g0, g1, g2, g3, 0);
#endif
    }
#else
    // ---- fallback: manual coalesced copy into the same padded layout
    #pragma unroll 1
    for (int it = 0; it < (O_*CC*24)/256; ++it) {
      int flat = it*256 + tid;
      int o    = flat / (CC*24);
      int rem  = flat - o*(CC*24);
      int cl   = rem / 24;
      int e    = (rem - cl*24) * 3;
      int jj   = e / 9;
      if (jj < validj) {
        int src = ((o*C_ + cbase + cl)*(OH_*OW_) + i*OW_ + j0 + jj)*9 + (e - jj*9);
        float w0 = Wg[src], w1 = Wg[src+1], w2 = Wg[src+2];
        int r = o*CC + cl;
        Wl[widx(r, e)]   = w0;
        Wl[widx(r, e+1)] = w1;
        Wl[widx(r, e+2)] = w2;
      }
    }
#endif

    // ---- stage x patch rows: x[b, cl, i..i+2, j0..j0+9]
    #pragma unroll 1
    for (int it = 0; it < (B_*CC*3*XCOL)/256; ++it) {  // 15 iterations
      int f   = it*256 + tid;
      int r   = f / XCOL;
      int col = f - r*XCOL;
      int b   = r / (CC*3);
      int rr  = r - b*(CC*3);
      int cl  = rr / 3;
      int u   = rr - cl*3;
      int gc  = j0 + col;
      float v = 0.f;
      if (gc < W_) v = Xg[((b*C_ + cbase + cl)*H_ + (i + u))*W_ + gc];
      Xl[f] = v;
    }

#if __has_builtin(__builtin_amdgcn_s_wait_tensorcnt)
    __builtin_amdgcn_s_wait_tensorcnt(0);   // no-op for waves that issued nothing
#endif
    __syncthreads();

    // ---- compute: each wave owns one j; 2x4 tiles of 16x16, K-step 4
    if (wid < validj) {
      #pragma unroll
      for (int step = 0; step < NSTEP; ++step) {
        const int kA = 4*step + 2*half;   // this lane's first K
        const int kB = kA + 1;
        const int clA = kA/9, rA = kA - clA*9, uA = rA/3, vA = rA - uA*3;
        const int clB = kB/9, rB = kB - clB*9, uB = rB/3, vB = rB - uB*3;
        const int offA = (clA*3 + uA)*XCOL + wid + vA;
        const int offB = (clB*3 + uB)*XCOL + wid + vB;
        const int eA = wid*9 + rA - uA*3 + uA*0;      // e for A-k not needed; B below
        (void)eA;

        v2f a[2];
        #pragma unroll
        for (int m = 0; m < 2; ++m) {
          const int brow = 16*m + row;
          a[m].x = Xl[brow*XBSTR + offA];
          a[m].y = Xl[brow*XBSTR + offB];
        }
        #pragma unroll
        for (int n = 0; n < 4; ++n) {
          const int o = 16*n + row;
          const int eBA = wid*9 + (kA - clA*9);
          const int eBB = wid*9 + (kB - clB*9);
          v2f bf;
          bf.x = Wl[widx(o*CC + clA, eBA)];
          bf.y = Wl[widx(o*CC + clB, eBB)];
          #pragma unroll
          for (int m = 0; m < 2; ++m)
            acc[m][n] = __builtin_amdgcn_wmma_f32_16x16x4_f32(
                false, a[m], false, bf, (short)0, acc[m][n], false, false);
        }
      }
    }
  }

  // ---- epilogue: bias + store
  if (wid < validj) {
    const int j = j0 + wid;
    #pragma unroll
    for (int n = 0; n < 4; ++n) {
      const int o = 16*n + row;
      const float bias = Bg[(o*OH_ + i)*OW_ + j];
      #pragma unroll
      for (int m = 0; m < 2; ++m) {
        #pragma unroll
        for (int g = 0; g < 8; ++g) {
          const int b = 16*m + 8*half + g;
          Og[((b*O_ + o)*OH_ + i)*OW_ + j] = acc[m][n][g] + bias;
        }
      }
    }
  }
}

extern "C" void kernel_launch(void* const* d_in, const int* in_sizes, int n_in,
                              void* d_out, int out_size, void* d_ws, size_t ws_size,
                              hipStream_t stream) {
  const float* x = (const float*)d_in[0];
  const float* w = (const float*)d_in[1];
  const float* b = (const float*)d_in[2];
  float* out = (float*)d_out;

  const size_t shmem = (size_t)(O_*CC*ROWW + B_*CC*3*XCOL) * sizeof(float); // 98304 B
  hipFuncSetAttribute(reinterpret_cast<const void*>(lcl_wmma_kernel),
                      hipFuncAttributeMaxDynamicSharedMemorySize, (int)shmem);

  dim3 grid((OW_ + TJ - 1)/TJ, OH_);   // 8 x 62 workgroups
  lcl_wmma_kernel<<<grid, 256, shmem, stream>>>(x, w, b, out);
}